// AbtCDR_60498909332002
// MI455X (gfx1250) — compile-verified
//
#include <hip/hip_runtime.h>
#include <cstdint>
#include <cstddef>

#define N_SRC_USERS 120000
#define N_TGT_USERS 100000
#define SH 4096
#define NUSERS (N_SRC_USERS + N_TGT_USERS - SH)   // 215904
#define N_SRC_ITEMS 60000
#define N_TGT_ITEMS 50000
#define NS (NUSERS + N_SRC_ITEMS)                 // 275904
#define NT (NUSERS + N_TGT_ITEMS)                 // 265904
#define NNZ_E 2000000

typedef __attribute__((ext_vector_type(2))) float v2f;
typedef __attribute__((ext_vector_type(8))) float v8f;

enum { EP_NONE = 0, EP_EXP = 1, EP_IDENT = 2, EP_SCALE_ADD = 3, EP_RELU = 4 };

#define LDP 68   // 272B row stride: 16B aligned, bank-spread

// ---------------------------------------------------------------------------
// Generic 64x64x64-tiled fp32 WMMA GEMM:  C[i,j] = sum_k Ae[i,k]*Be[j,k]
//   Ae[i,k] = transA ? A[k*lda+i] : A[i*lda+k]
//   Be[j,k] = transB ? B[k*ldb+j] : B[j*ldb+k]
// 256 threads = 8 waves; wave w owns rows 16*(w>>1), cols 32*(w&1) of tile.
// ---------------------------------------------------------------------------
__global__ __launch_bounds__(256) void k_gemm(
    const float* __restrict__ A, const float* __restrict__ B, float* __restrict__ C,
    int M, int N, int K, int lda, int ldb, int ldc, int transA, int transB,
    int ep, float alpha, const float* __restrict__ rowscale,
    const float* __restrict__ Cadd, int ldca)
{
  __shared__ float As[64][LDP];
  __shared__ float Bs[64][LDP];
  const int tid  = threadIdx.x;
  const int wave = tid >> 5, lane = tid & 31;
  const int row0 = blockIdx.y * 64, col0 = blockIdx.x * 64;
  const int wr = (wave >> 1) * 16, wc = (wave & 1) * 32;
  const int lrow = tid >> 2, lq = (tid & 3) * 16;
  const int m = lane & 15, koff = (lane >> 4) * 2;

  v8f c0 = {}; v8f c1 = {};

  for (int kk = 0; kk < K; kk += 64) {
    if (!transA) {
      const float* ap = A + (size_t)(row0 + lrow) * lda + kk + lq;
      #pragma unroll
      for (int i = 0; i < 4; ++i)
        *(float4*)&As[lrow][lq + 4 * i] = *(const float4*)(ap + 4 * i);
    } else {
      const float* ap = A + (size_t)(kk + lrow) * lda + row0 + lq;
      #pragma unroll
      for (int i = 0; i < 4; ++i) {
        float4 t = *(const float4*)(ap + 4 * i);
        As[lq + 4*i + 0][lrow] = t.x;  As[lq + 4*i + 1][lrow] = t.y;
        As[lq + 4*i + 2][lrow] = t.z;  As[lq + 4*i + 3][lrow] = t.w;
      }
    }
    if (!transB) {
      const float* bp = B + (size_t)(col0 + lrow) * ldb + kk + lq;
      #pragma unroll
      for (int i = 0; i < 4; ++i)
        *(float4*)&Bs[lrow][lq + 4 * i] = *(const float4*)(bp + 4 * i);
    } else {
      const float* bp = B + (size_t)(kk + lrow) * ldb + col0 + lq;
      #pragma unroll
      for (int i = 0; i < 4; ++i) {
        float4 t = *(const float4*)(bp + 4 * i);
        Bs[lq + 4*i + 0][lrow] = t.x;  Bs[lq + 4*i + 1][lrow] = t.y;
        Bs[lq + 4*i + 2][lrow] = t.z;  Bs[lq + 4*i + 3][lrow] = t.w;
      }
    }
    __syncthreads();
    #pragma unroll
    for (int k0 = 0; k0 < 64; k0 += 4) {
      v2f a  = *(const v2f*)&As[wr + m][k0 + koff];
      v2f b0 = *(const v2f*)&Bs[wc + m][k0 + koff];
      v2f b1 = *(const v2f*)&Bs[wc + 16 + m][k0 + koff];
      c0 = __builtin_amdgcn_wmma_f32_16x16x4_f32(false, a, false, b0, (short)0, c0, false, false);
      c1 = __builtin_amdgcn_wmma_f32_16x16x4_f32(false, a, false, b1, (short)0, c1, false, false);
    }
    __syncthreads();
  }

  // C/D layout: VGPR v: lanes0-15 -> M=v, lanes16-31 -> M=v+8; N = lane&15
  const int cn = lane & 15;
  const int rb = (lane >> 4) * 8;
  #pragma unroll
  for (int v = 0; v < 8; ++v) {
    const int gr  = row0 + wr + rb + v;
    const int gc0 = col0 + wc + cn;
    const int gc1 = gc0 + 16;
    float x0 = c0[v], x1 = c1[v];
    if (ep == EP_EXP)        { x0 = expf(x0 * alpha); x1 = expf(x1 * alpha); }
    else if (ep == EP_IDENT) { x0 += (gr == gc0) ? 1.0f : 0.0f;
                               x1 += (gr == gc1) ? 1.0f : 0.0f; }
    else if (ep == EP_SCALE_ADD) {
      float s = rowscale[gr];
      x0 = Cadd[(size_t)gr * ldca + gc0] + x0 * s;
      x1 = Cadd[(size_t)gr * ldca + gc1] + x1 * s;
    }
    else if (ep == EP_RELU)  { x0 = fmaxf(x0, 0.0f); x1 = fmaxf(x1, 0.0f); }
    C[(size_t)gr * ldc + gc0] = x0;
    C[(size_t)gr * ldc + gc1] = x1;
  }
}

// ---------------------------------------------------------------------------
// SpMM: out[idx0[e],:] += vals[e] * x[idx1[e],:]  (D=64, wave per edge)
// ---------------------------------------------------------------------------
__global__ __launch_bounds__(256) void k_spmm(
    const int* __restrict__ idx, const float* __restrict__ vals,
    const float* __restrict__ x, float* __restrict__ out, int nnz)
{
  int e = blockIdx.x * 8 + (threadIdx.x >> 5);
  if (e >= nnz) return;
  int lane = threadIdx.x & 31;
  int r = idx[e];
  int c = idx[nnz + e];
  float v = vals[e];
  float2 xv = *(const float2*)(x + (size_t)c * 64 + lane * 2);
  float* op = out + (size_t)r * 64 + lane * 2;
  unsafeAtomicAdd(op,     v * xv.x);
  unsafeAtomicAdd(op + 1, v * xv.y);
}

// init: emb = concat(user,item); also write layer-0 (raw) columns of d_out
__global__ __launch_bounds__(256) void k_init(
    const float* __restrict__ user, const float* __restrict__ item,
    int n_user_el, int n_total_el, float* __restrict__ emb,
    float* __restrict__ dout, long long dout_row_off)
{
  int i = blockIdx.x * 256 + threadIdx.x;
  if (i >= n_total_el) return;
  float v = (i < n_user_el) ? user[i] : item[i - n_user_el];
  emb[i] = v;
  int r = i >> 6, c = i & 63;
  dout[(size_t)(dout_row_off + r) * 256 + c] = v;
}

__global__ __launch_bounds__(256) void k_rowsum_inv(
    const float* __restrict__ s, float* __restrict__ inv, int n)
{
  __shared__ float red[256];
  const float* p = s + (size_t)blockIdx.x * n;
  float acc = 0.0f;
  for (int j = threadIdx.x; j < n; j += 256) acc += p[j];
  red[threadIdx.x] = acc; __syncthreads();
  for (int st = 128; st > 0; st >>= 1) {
    if (threadIdx.x < st) red[threadIdx.x] += red[threadIdx.x + st];
    __syncthreads();
  }
  if (threadIdx.x == 0) inv[blockIdx.x] = 1.0f / fmaxf(red[0], 1e-12f);
}

__global__ __launch_bounds__(256) void k_colsum_inv(
    const float* __restrict__ s, float* __restrict__ inv, int n)
{
  int j = blockIdx.x * 256 + threadIdx.x;
  if (j >= n) return;
  float acc = 0.0f;
  for (int i = 0; i < n; ++i) acc += s[(size_t)i * n + j];
  inv[j] = 1.0f / fmaxf(acc, 1e-12f);
}

// l2-normalize rows of X[rows,64] -> Y (row stride ldy, column offset coloff)
__global__ __launch_bounds__(256) void k_l2n(
    const float* __restrict__ X, float* __restrict__ Y,
    int rows, long long ldy, int coloff)
{
  int row = blockIdx.x * 8 + (threadIdx.x >> 5);
  if (row >= rows) return;
  int lane = threadIdx.x & 31;
  float2 v = *(const float2*)(X + (size_t)row * 64 + lane * 2);
  float ss = v.x * v.x + v.y * v.y;
  #pragma unroll
  for (int msk = 16; msk > 0; msk >>= 1) ss += __shfl_xor(ss, msk, 32);
  float s = 1.0f / fmaxf(sqrtf(ss), 1e-12f);
  float2 o; o.x = v.x * s; o.y = v.y * s;
  *(float2*)(Y + (size_t)row * ldy + coloff + lane * 2) = o;
}

__global__ __launch_bounds__(256) void k_combine(
    float* __restrict__ emb, const float* __restrict__ prop, int n)
{
  int i = blockIdx.x * 256 + threadIdx.x;
  if (i < n) emb[i] = (emb[i] + prop[i]) * 0.5f;
}

// ---------------------------------------------------------------------------
static void gemm(hipStream_t st, const float* A, const float* B, float* C,
                 int M, int N, int K, int lda, int ldb, int ldc,
                 int tA, int tB, int ep, float alpha,
                 const float* rs, const float* Cadd, int ldca)
{
  dim3 g(N / 64, M / 64), b(256);
  k_gemm<<<g, b, 0, st>>>(A, B, C, M, N, K, lda, ldb, ldc, tA, tB, ep, alpha, rs, Cadd, ldca);
}

extern "C" void kernel_launch(void* const* d_in, const int* in_sizes, int n_in,
                              void* d_out, int out_size, void* d_ws, size_t ws_size,
                              hipStream_t stream)
{
  const float* src_user   = (const float*)d_in[0];
  const float* tgt_user   = (const float*)d_in[1];
  const float* src_item   = (const float*)d_in[2];
  const float* tgt_item   = (const float*)d_in[3];
  const float* mapping    = (const float*)d_in[4];
  const float* adj_s_vals = (const float*)d_in[5];
  const float* adj_t_vals = (const float*)d_in[6];
  const int*   adj_s_idx  = (const int*)d_in[7];
  const int*   adj_t_idx  = (const int*)d_in[8];
  float* out = (float*)d_out;

  float* w = (float*)d_ws;
  size_t off = 0;
  auto alloc = [&](size_t n) { float* p = w + off; off += n; return p; };
  float* embS[2] = { alloc((size_t)NS * 64), alloc((size_t)NS * 64) };
  float* embT[2] = { alloc((size_t)NT * 64), alloc((size_t)NT * 64) };
  float* smat = alloc((size_t)SH * SH);
  float* adjS = alloc((size_t)SH * SH);
  float* adjT = alloc((size_t)SH * SH);
  float* m1   = alloc((size_t)SH * 64);
  float* srcU = alloc((size_t)SH * 64);
  float* tgtU = alloc((size_t)SH * 64);
  float* pA   = alloc((size_t)SH * 64);
  float* pB   = alloc((size_t)SH * 64);
  float* qA   = alloc((size_t)SH * 64);
  float* qB   = alloc((size_t)SH * 64);
  float* invR = alloc(SH);
  float* invC = alloc(SH);

  // layer 0: build embeddings + raw columns 0..63 of d_out
  k_init<<<((size_t)NS * 64 + 255) / 256, 256, 0, stream>>>(
      src_user, src_item, NUSERS * 64, NS * 64, embS[0], out, 0);
  k_init<<<((size_t)NT * 64 + 255) / 256, 256, 0, stream>>>(
      tgt_user, tgt_item, NUSERS * 64, NT * 64, embT[0], out, (long long)NS);

  int cur = 0;
  for (int layer = 0; layer < 3; ++layer) {
    int nxt = cur ^ 1;
    hipMemsetAsync(embS[nxt], 0, (size_t)NS * 64 * sizeof(float), stream);
    hipMemsetAsync(embT[nxt], 0, (size_t)NT * 64 * sizeof(float), stream);
    k_spmm<<<(NNZ_E + 7) / 8, 256, 0, stream>>>(adj_s_idx, adj_s_vals, embS[cur], embS[nxt], NNZ_E);
    k_spmm<<<(NNZ_E + 7) / 8, 256, 0, stream>>>(adj_t_idx, adj_t_vals, embT[cur], embT[nxt], NNZ_E);

    if (layer == 0) {
      const float* sraw = embS[nxt];   // first SH rows = src_raw
      const float* traw = embT[nxt];   // first SH rows = tgt_raw
      // m1 = src_raw @ mapping      (NN -> transB on mapping)
      gemm(stream, sraw, mapping, m1, SH, 64, 64, 64, 64, 64, 0, 1, EP_NONE, 0.f, nullptr, nullptr, 0);
      // s = exp(m1 @ tgt_raw^T / T) (NT)
      gemm(stream, m1, traw, smat, SH, SH, 64, 64, 64, SH, 0, 0, EP_EXP, 0.2f, nullptr, nullptr, 0);
      k_rowsum_inv<<<SH, 256, 0, stream>>>(smat, invR, SH);
      k_colsum_inv<<<SH / 256, 256, 0, stream>>>(smat, invC, SH);
      // src' = src_raw + diag(invR) * (s @ tgt_raw)
      gemm(stream, smat, traw, srcU, SH, 64, SH, SH, 64, 64, 0, 1, EP_SCALE_ADD, 0.f, invR, sraw, 64);
      // tgt' = tgt_raw + diag(invC) * (s^T @ src')
      gemm(stream, smat, srcU, tgtU, SH, 64, SH, SH, 64, 64, 1, 1, EP_SCALE_ADD, 0.f, invC, traw, 64);
      // adj_s = s@s^T + I ; adj_t = s^T@s + I   (row-norm skipped: relu/l2n invariant)
      gemm(stream, smat, smat, adjS, SH, SH, SH, SH, SH, SH, 0, 0, EP_IDENT, 0.f, nullptr, nullptr, 0);
      gemm(stream, smat, smat, adjT, SH, SH, SH, SH, SH, SH, 1, 1, EP_IDENT, 0.f, nullptr, nullptr, 0);
      // 3x propagation + l2n, src side
      gemm(stream, adjS, srcU, pA, SH, 64, SH, SH, 64, 64, 0, 1, EP_RELU, 0.f, nullptr, nullptr, 0);
      k_l2n<<<SH / 8, 256, 0, stream>>>(pA, pA, SH, 64, 0);
      gemm(stream, adjS, pA, pB, SH, 64, SH, SH, 64, 64, 0, 1, EP_RELU, 0.f, nullptr, nullptr, 0);
      k_l2n<<<SH / 8, 256, 0, stream>>>(pB, pB, SH, 64, 0);
      gemm(stream, adjS, pB, pA, SH, 64, SH, SH, 64, 64, 0, 1, EP_RELU, 0.f, nullptr, nullptr, 0);
      k_l2n<<<SH / 8, 256, 0, stream>>>(pA, pA, SH, 64, 0);
      // tgt side
      gemm(stream, adjT, tgtU, qA, SH, 64, SH, SH, 64, 64, 0, 1, EP_RELU, 0.f, nullptr, nullptr, 0);
      k_l2n<<<SH / 8, 256, 0, stream>>>(qA, qA, SH, 64, 0);
      gemm(stream, adjT, qA, qB, SH, 64, SH, SH, 64, 64, 0, 1, EP_RELU, 0.f, nullptr, nullptr, 0);
      k_l2n<<<SH / 8, 256, 0, stream>>>(qB, qB, SH, 64, 0);
      gemm(stream, adjT, qB, qA, SH, 64, SH, SH, 64, 64, 0, 1, EP_RELU, 0.f, nullptr, nullptr, 0);
      k_l2n<<<SH / 8, 256, 0, stream>>>(qA, qA, SH, 64, 0);
      // writeback shared block: emb = (raw + prop) * 0.5
      k_combine<<<(SH * 64 + 255) / 256, 256, 0, stream>>>(embS[nxt], pA, SH * 64);
      k_combine<<<(SH * 64 + 255) / 256, 256, 0, stream>>>(embT[nxt], qA, SH * 64);
    }

    // list append: l2n(emb) into d_out columns 64*(layer+1)..
    k_l2n<<<(NS + 7) / 8, 256, 0, stream>>>(embS[nxt], out, NS, 256, 64 * (layer + 1));
    k_l2n<<<(NT + 7) / 8, 256, 0, stream>>>(embT[nxt], out + (size_t)NS * 256, NT, 256, 64 * (layer + 1));
    cur = nxt;
  }
}